// VGAE_11218454577549
// MI455X (gfx1250) — compile-verified
//
#include <hip/hip_runtime.h>

typedef __attribute__((ext_vector_type(2))) float v2f;
typedef __attribute__((ext_vector_type(8))) float v8f;

#define Nn 10000
#define Ee 160000
#define DIN 512
#define DH 256
#define DZ 64

// ---------------------------------------------------------------- utilities
__global__ __launch_bounds__(256) void fill_f32(float* __restrict__ p, float v, int n) {
  int i = blockIdx.x * blockDim.x + threadIdx.x;
  if (i < n) p[i] = v;
}

// deg[dst] += 1 for each edge (deg pre-filled with 1.0 for the self loop)
__global__ __launch_bounds__(256) void deg_kernel(const int* __restrict__ ei,
                                                  float* __restrict__ deg, int E) {
  int e = blockIdx.x * blockDim.x + threadIdx.x;
  if (e < E) atomicAdd(&deg[ei[E + e]], 1.0f);
}

__global__ __launch_bounds__(256) void rsqrt_kernel(const float* __restrict__ deg,
                                                    float* __restrict__ dinv, int n) {
  int i = blockIdx.x * blockDim.x + threadIdx.x;
  if (i < n) dinv[i] = rsqrtf(deg[i]);
}

// ------------------------------------------------------- WMMA GEMM (4x4 macro)
// C[M,N] = A[M,K] @ B[K,N], row-major fp32.  One wave owns a 64x64 macro-tile
// (4x4 grid of v_wmma_f32_16x16x4_f32 tiles -> 16 independent acc chains).
// Operand rows/cols are clamped for ragged macro-tiles (loads harmless, EXEC
// stays all-ones); stores are guarded per 16-tile with wave-uniform predicates.
__global__ __launch_bounds__(256) void gemm_wmma_f32_4x4(
    const float* __restrict__ A, const float* __restrict__ B, float* __restrict__ C,
    int Mtiles, int Ntiles, int K, int lda, int ldb, int ldc) {
  int Mg = (Mtiles + 3) >> 2;
  int Ng = (Ntiles + 3) >> 2;
  int wave = blockIdx.x * (blockDim.x >> 5) + (threadIdx.x >> 5);
  if (wave >= Mg * Ng) return;  // wave-uniform
  int mg = wave / Ng;
  int ng = wave % Ng;
  int lane = threadIdx.x & 31;
  int half = lane >> 4;   // selects K pair {0,1} vs {2,3} within each k-step
  int r16  = lane & 15;
  int M = Mtiles * 16, N = Ntiles * 16;

  const float* ap[4];
  const float* bp[4];
#pragma unroll
  for (int i = 0; i < 4; ++i) {
    int row = mg * 64 + i * 16 + r16;
    if (row >= M) row = M - 1;                         // clamp (load-safe)
    ap[i] = A + (size_t)row * lda + 2 * half;          // A[row][k + 2h .. +1]
  }
#pragma unroll
  for (int j = 0; j < 4; ++j) {
    int col = ng * 64 + j * 16 + r16;
    if (col >= N) col = N - 1;
    bp[j] = B + (size_t)(2 * half) * ldb + col;        // B[k + 2h][col]
  }

  v8f acc[16];
  v8f zero = {};
#pragma unroll
  for (int t = 0; t < 16; ++t) acc[t] = zero;

  for (int k = 0; k < K; k += 4) {
    v2f a[4], b[4];
#pragma unroll
    for (int i = 0; i < 4; ++i) a[i] = *(const v2f*)(ap[i] + k);
#pragma unroll
    for (int j = 0; j < 4; ++j) {
      b[j].x = bp[j][(size_t)k * ldb];
      b[j].y = bp[j][(size_t)k * ldb + ldb];
    }
#pragma unroll
    for (int i = 0; i < 4; ++i)
#pragma unroll
      for (int j = 0; j < 4; ++j)
        acc[i * 4 + j] = __builtin_amdgcn_wmma_f32_16x16x4_f32(
            false, a[i], false, b[j], (short)0, acc[i * 4 + j], false, false);
  }

#pragma unroll
  for (int i = 0; i < 4; ++i) {
#pragma unroll
    for (int j = 0; j < 4; ++j) {
      int mt = mg * 4 + i, nt = ng * 4 + j;
      if (mt < Mtiles && nt < Ntiles) {                // wave-uniform guard
        float* cp = C + (size_t)(mt * 16 + half * 8) * ldc + nt * 16 + r16;
#pragma unroll
        for (int r = 0; r < 8; ++r) cp[(size_t)r * ldc] = acc[i * 4 + j][r];
      }
    }
  }
}

// ------------------------------------------------------- WMMA Gram (4x4 macro)
// C[M,M] = Z[M,64] @ Z[M,64]^T.  B-tile addressing == A-tile addressing with the
// column-block base, so both operands are contiguous float2 loads (Z = 2.56 MB,
// fully L2-resident).  64x64 macro-tile per wave; K=64 fully unrolled.
__global__ __launch_bounds__(256) void gram_wmma_f32_4x4(const float* __restrict__ Z,
                                                         float* __restrict__ C,
                                                         int Mtiles) {
  int Gg = (Mtiles + 3) >> 2;                           // 157 for 625 tiles
  int wave = blockIdx.x * (blockDim.x >> 5) + (threadIdx.x >> 5);
  if (wave >= Gg * Gg) return;
  int mg = wave / Gg;
  int ng = wave % Gg;
  int lane = threadIdx.x & 31;
  int half = lane >> 4;
  int r16  = lane & 15;
  int M = Mtiles * 16;

  const float* ap[4];
  const float* bp[4];
#pragma unroll
  for (int i = 0; i < 4; ++i) {
    int row = mg * 64 + i * 16 + r16;
    if (row >= M) row = M - 1;
    ap[i] = Z + (size_t)row * DZ + 2 * half;
  }
#pragma unroll
  for (int j = 0; j < 4; ++j) {
    int col = ng * 64 + j * 16 + r16;
    if (col >= M) col = M - 1;
    bp[j] = Z + (size_t)col * DZ + 2 * half;
  }

  v8f acc[16];
  v8f zero = {};
#pragma unroll
  for (int t = 0; t < 16; ++t) acc[t] = zero;

#pragma unroll
  for (int s = 0; s < 16; ++s) {                        // K = 64, 4 per step
    v2f a[4], b[4];
#pragma unroll
    for (int i = 0; i < 4; ++i) a[i] = *(const v2f*)(ap[i] + 4 * s);
#pragma unroll
    for (int j = 0; j < 4; ++j) b[j] = *(const v2f*)(bp[j] + 4 * s);
#pragma unroll
    for (int i = 0; i < 4; ++i)
#pragma unroll
      for (int j = 0; j < 4; ++j)
        acc[i * 4 + j] = __builtin_amdgcn_wmma_f32_16x16x4_f32(
            false, a[i], false, b[j], (short)0, acc[i * 4 + j], false, false);
  }

#pragma unroll
  for (int i = 0; i < 4; ++i) {
#pragma unroll
    for (int j = 0; j < 4; ++j) {
      int mt = mg * 4 + i, nt = ng * 4 + j;
      if (mt < Mtiles && nt < Mtiles) {
        float* cp = C + (size_t)(mt * 16 + half * 8) * Nn + nt * 16 + r16;
#pragma unroll
        for (int r = 0; r < 8; ++r) cp[(size_t)r * Nn] = acc[i * 4 + j][r];
      }
    }
  }
}

// ---------------------------------------------------------------- scatters
// agg[dst, f] += h[src, f] * dinv[src]*dinv[dst], DH=256 features, 1 block/edge
__global__ __launch_bounds__(256) void scatter_dh(const int* __restrict__ ei,
                                                  const float* __restrict__ dinv,
                                                  const float* __restrict__ h,
                                                  float* __restrict__ agg, int E) {
  int e = blockIdx.x;
  int f = threadIdx.x;
  int s = ei[e], d = ei[E + e];
  float norm = dinv[s] * dinv[d];
  atomicAdd(&agg[(size_t)d * DH + f], h[(size_t)s * DH + f] * norm);
}

// two DZ=64 scatters (mu and logstd) per edge, 1 block of 128 threads per edge
__global__ __launch_bounds__(128) void scatter_dz2(const int* __restrict__ ei,
                                                   const float* __restrict__ dinv,
                                                   const float* __restrict__ hmu,
                                                   const float* __restrict__ hls,
                                                   float* __restrict__ aggmu,
                                                   float* __restrict__ aggls, int E) {
  int e = blockIdx.x;
  int t = threadIdx.x;
  int s = ei[e], d = ei[E + e];
  float norm = dinv[s] * dinv[d];
  if (t < DZ) {
    atomicAdd(&aggmu[(size_t)d * DZ + t], hmu[(size_t)s * DZ + t] * norm);
  } else {
    int f = t - DZ;
    atomicAdd(&aggls[(size_t)d * DZ + f], hls[(size_t)s * DZ + f] * norm);
  }
}

// ---------------------------------------------------------------- finalizers
// hidden = relu(agg + h1/deg + b1)      (1/deg == dinv^2)
__global__ __launch_bounds__(256) void finalize_hidden(
    const float* __restrict__ agg, const float* __restrict__ h1,
    const float* __restrict__ dinv, const float* __restrict__ b1,
    float* __restrict__ hidden, int n) {
  int i = blockIdx.x * blockDim.x + threadIdx.x;
  if (i >= n) return;
  int node = i >> 8;   // DH = 256
  int f    = i & 255;
  float di = dinv[node];
  float v = agg[i] + h1[i] * di * di + b1[f];
  hidden[i] = v > 0.0f ? v : 0.0f;
}

// mu/logstd finalization + reparameterization z = mu + eps*exp(logstd)
__global__ __launch_bounds__(256) void finalize_z(
    const float* __restrict__ aggmu, const float* __restrict__ aggls,
    const float* __restrict__ mulin, const float* __restrict__ lslin,
    const float* __restrict__ dinv, const float* __restrict__ bmu,
    const float* __restrict__ bls, const float* __restrict__ eps,
    float* __restrict__ omu, float* __restrict__ ols, float* __restrict__ z, int n) {
  int i = blockIdx.x * blockDim.x + threadIdx.x;
  if (i >= n) return;
  int node = i >> 6;   // DZ = 64
  int f    = i & 63;
  float di = dinv[node];
  float di2 = di * di;
  float mu = aggmu[i] + mulin[i] * di2 + bmu[f];
  float ls = aggls[i] + lslin[i] * di2 + bls[f];
  omu[i] = mu;
  ols[i] = ls;
  z[i] = mu + eps[i] * expf(ls);
}

// ---------------------------------------------------------------- launcher
extern "C" void kernel_launch(void* const* d_in, const int* in_sizes, int n_in,
                              void* d_out, int out_size, void* d_ws, size_t ws_size,
                              hipStream_t stream) {
  const float* x   = (const float*)d_in[0];
  const int*   ei  = (const int*)d_in[1];
  const float* eps = (const float*)d_in[2];
  const float* W1  = (const float*)d_in[3];
  const float* b1  = (const float*)d_in[4];
  const float* Wmu = (const float*)d_in[5];
  const float* bmu = (const float*)d_in[6];
  const float* Wls = (const float*)d_in[7];
  const float* bls = (const float*)d_in[8];

  float* out = (float*)d_out;
  float* adj = out;                              // [N, N]
  float* omu = out + (size_t)Nn * Nn;            // [N, DZ]
  float* ols = omu + (size_t)Nn * DZ;            // [N, DZ]

  float* ws = (float*)d_ws;
  float* deg    = ws; ws += 10240;               // padded for alignment
  float* dinv   = ws; ws += 10240;
  float* h1     = ws; ws += (size_t)Nn * DH;
  float* agg1   = ws; ws += (size_t)Nn * DH;
  float* hidden = ws; ws += (size_t)Nn * DH;
  float* mulin  = ws; ws += (size_t)Nn * DZ;
  float* lslin  = ws; ws += (size_t)Nn * DZ;
  float* aggmu  = ws; ws += (size_t)Nn * DZ;
  float* aggls  = ws; ws += (size_t)Nn * DZ;
  float* zbuf   = ws; ws += (size_t)Nn * DZ;

  const int B = 256;
  // init accumulators
  fill_f32<<<(Nn + B - 1) / B, B, 0, stream>>>(deg, 1.0f, Nn);            // self-loop
  fill_f32<<<((Nn * DH) + B - 1) / B, B, 0, stream>>>(agg1, 0.0f, Nn * DH);
  fill_f32<<<((Nn * DZ) + B - 1) / B, B, 0, stream>>>(aggmu, 0.0f, Nn * DZ);
  fill_f32<<<((Nn * DZ) + B - 1) / B, B, 0, stream>>>(aggls, 0.0f, Nn * DZ);

  // degree + normalization
  deg_kernel<<<(Ee + B - 1) / B, B, 0, stream>>>(ei, deg, Ee);
  rsqrt_kernel<<<(Nn + B - 1) / B, B, 0, stream>>>(deg, dinv, Nn);

  // layer 1: h1 = x @ W1   (625x16 tiles -> 157x4 macro-groups)
  {
    int mt = Nn / 16, nt = DH / 16;
    int waves = ((mt + 3) / 4) * ((nt + 3) / 4);
    gemm_wmma_f32_4x4<<<(waves + 7) / 8, 256, 0, stream>>>(x, W1, h1, mt, nt, DIN,
                                                           DIN, DH, DH);
  }
  scatter_dh<<<Ee, DH, 0, stream>>>(ei, dinv, h1, agg1, Ee);
  finalize_hidden<<<((Nn * DH) + B - 1) / B, B, 0, stream>>>(agg1, h1, dinv, b1,
                                                             hidden, Nn * DH);

  // layer 2: mu/logstd linear parts (625x4 tiles -> 157x1 macro-groups)
  {
    int mt = Nn / 16, nt = DZ / 16;
    int waves = ((mt + 3) / 4) * ((nt + 3) / 4);
    gemm_wmma_f32_4x4<<<(waves + 7) / 8, 256, 0, stream>>>(hidden, Wmu, mulin, mt,
                                                           nt, DH, DH, DZ, DZ);
    gemm_wmma_f32_4x4<<<(waves + 7) / 8, 256, 0, stream>>>(hidden, Wls, lslin, mt,
                                                           nt, DH, DH, DZ, DZ);
  }
  scatter_dz2<<<Ee, 128, 0, stream>>>(ei, dinv, mulin, lslin, aggmu, aggls, Ee);
  finalize_z<<<((Nn * DZ) + B - 1) / B, B, 0, stream>>>(aggmu, aggls, mulin, lslin,
                                                        dinv, bmu, bls, eps, omu,
                                                        ols, zbuf, Nn * DZ);

  // decoder: adj = z @ z^T  (625x625 tiles -> 157x157 macro-groups)
  {
    int mt = Nn / 16;
    long long waves = (long long)((mt + 3) / 4) * ((mt + 3) / 4);
    gram_wmma_f32_4x4<<<(unsigned)((waves + 7) / 8), 256, 0, stream>>>(zbuf, adj, mt);
  }
}